// YuaTopKRouter_61881888800981
// MI455X (gfx1250) — compile-verified
//
#include <hip/hip_runtime.h>
#include <hip/hip_bf16.h>

typedef float v2f __attribute__((ext_vector_type(2)));
typedef float v8f __attribute__((ext_vector_type(8)));

#define TOKENS   32768
#define HID      4096
#define NEXP     64
#define WAVES_PB 8            // 256 threads / wave32
#define ROWSTR   65           // padded LDS row stride (floats) to avoid bank conflicts

// One wave computes a 16-token x 64-expert logit tile via V_WMMA_F32_16X16X4_F32,
// then does top-2 + softmax for its 16 tokens.
__global__ __launch_bounds__(256)
void YuaTopKRouter_kernel(const float* __restrict__ H,
                          const float* __restrict__ W,
                          float* __restrict__ out_w,   // [TOKENS,2] weights
                          int*   __restrict__ out_i)   // [TOKENS,2] indices
{
    __shared__ float lds[WAVES_PB * 16 * ROWSTR];

    const int tid  = threadIdx.x;
    const int lane = tid & 31;
    const int wave = tid >> 5;
    const int l16  = lane & 15;
    const int hi   = lane >> 4;            // 0: lanes 0-15, 1: lanes 16-31

    const int m0 = (blockIdx.x * WAVES_PB + wave) * 16;   // first token of this wave's tile

    // A fragment (16x4, f32): lane l16 carries token row m0+l16.
    //   VGPR0 = K = k + 2*hi, VGPR1 = K = k + 2*hi + 1  -> one float2 load.
    const float* arow = H + (size_t)(m0 + l16) * HID + 2 * hi;
    // B fragment (4x16): B[k][n] = W[n][k]; lane l16 carries expert column (nt*16 + l16).
    const float* brow = W + (size_t)l16 * HID + 2 * hi;

    v8f acc0 = {}, acc1 = {}, acc2 = {}, acc3 = {};

#pragma unroll 4
    for (int k = 0; k < HID; k += 4) {
        v2f a  = *(const v2f*)(arow + k);
        v2f b0 = *(const v2f*)(brow + k);
        v2f b1 = *(const v2f*)(brow + 16 * HID + k);
        v2f b2 = *(const v2f*)(brow + 32 * HID + k);
        v2f b3 = *(const v2f*)(brow + 48 * HID + k);
        acc0 = __builtin_amdgcn_wmma_f32_16x16x4_f32(false, a, false, b0, (short)0, acc0, false, false);
        acc1 = __builtin_amdgcn_wmma_f32_16x16x4_f32(false, a, false, b1, (short)0, acc1, false, false);
        acc2 = __builtin_amdgcn_wmma_f32_16x16x4_f32(false, a, false, b2, (short)0, acc2, false, false);
        acc3 = __builtin_amdgcn_wmma_f32_16x16x4_f32(false, a, false, b3, (short)0, acc3, false, false);
    }

    // Spill logits to LDS.  C/D layout: VGPR r, lanes<16 -> row M=r, lanes>=16 -> row M=r+8;
    // column N = l16.  token = r + 8*hi, expert = 16*tile + l16.
    float* wlds = lds + wave * 16 * ROWSTR;
#pragma unroll
    for (int r = 0; r < 8; ++r) {
        const int trow = (r + 8 * hi) * ROWSTR + l16;
        wlds[trow +  0] = acc0[r];
        wlds[trow + 16] = acc1[r];
        wlds[trow + 32] = acc2[r];
        wlds[trow + 48] = acc3[r];
    }
    __syncthreads();

    // Lanes 0-15: one token each -> top-2 over 64 experts, softmax over the pair.
    if (lane < 16) {
        const float* row = wlds + l16 * ROWSTR;
        float m1 = -__builtin_inff(); int i1 = 0;
        float m2 = -__builtin_inff(); int i2 = 0;
#pragma unroll
        for (int e = 0; e < NEXP; ++e) {
            float v = row[e];
            if (v > m1)      { m2 = m1; i2 = i1; m1 = v; i1 = e; }
            else if (v > m2) { m2 = v;  i2 = e; }
        }
        // softmax over [m1, m2] with m1 >= m2
        float ex = __expf(m2 - m1);
        float w1 = 1.0f / (1.0f + ex);
        float w2 = 1.0f - w1;

        const int tok = m0 + l16;
        out_w[2 * tok + 0] = w1;
        out_w[2 * tok + 1] = w2;
        out_i[2 * tok + 0] = i1;
        out_i[2 * tok + 1] = i2;
    }
}

extern "C" void kernel_launch(void* const* d_in, const int* in_sizes, int n_in,
                              void* d_out, int out_size, void* d_ws, size_t ws_size,
                              hipStream_t stream) {
    (void)in_sizes; (void)n_in; (void)d_ws; (void)ws_size; (void)out_size;
    const float* H = (const float*)d_in[0];     // [32768, 4096] f32
    const float* W = (const float*)d_in[1];     // [64, 4096]    f32
    float* out_w = (float*)d_out;               // first output: weights [32768,2]
    int*   out_i = (int*)((float*)d_out + 2 * TOKENS);  // second output: indices [32768,2]

    dim3 grid(TOKENS / (WAVES_PB * 16));        // 256 workgroups
    dim3 block(256);                            // 8 wave32s
    YuaTopKRouter_kernel<<<grid, block, 0, stream>>>(H, W, out_w, out_i);
}